// LengthRegulator_38259568673385
// MI455X (gfx1250) — compile-verified
//
#include <hip/hip_runtime.h>

// LengthRegulator for MI455X (gfx1250).
// Pure gather, 0 FLOPs, ~75 MB HBM traffic -> memory-bound (~3.2 us at 23.3 TB/s).
// Uses the CDNA5 async global<->LDS DMA path (ASYNCcnt) for the row copies:
// global_load_async_to_lds_b128 -> s_wait_asynccnt -> global_store_async_from_lds_b128.

#define B_ 32
#define L_ 512
#define D_ 256

#define FPW 4            // frames per wave
#define T_TILE 32        // 8 waves * FPW frames per block

// Builtin signature (from hipcc diagnostic): params are pointers to a 16-byte
// int vector, global side in AS1, LDS side in AS3.
typedef int v4i __attribute__((vector_size(16)));
typedef __attribute__((address_space(1))) v4i* gv4i_p;   // global
typedef __attribute__((address_space(3))) v4i* lv4i_p;   // LDS

// ---------------------------------------------------------------------------
// Gather kernel (placed first so the disasm snippet shows the async path).
// Grid (ceil(max_T/T_TILE), B), 256 threads = 8 waves; wave w owns frames
// tileBase + w + 8*k, k=0..3.
// Phase 1: binary search + async global->LDS loads for all gather frames.
// One s_wait_asynccnt. Phase 2: async LDS->global stores (zeros for padding).
// ---------------------------------------------------------------------------
__global__ __launch_bounds__(256) void lr_gather_kernel(
    const float* __restrict__ hidden, const int* __restrict__ cum,
    const int* __restrict__ totals, float* __restrict__ out, int max_T) {
  __shared__ int cumS[L_];                    // 2 KB
  __shared__ float4 stage[8][FPW][64];        // 8 waves * 4 frames * 1 KB = 32 KB

  const int b = blockIdx.y;
  const int tileBase = blockIdx.x * T_TILE;
  const int tid = threadIdx.x;
  const int w = tid >> 5;
  const int lane = tid & 31;

  // Stage this batch's cumulative-duration row in LDS (coalesced, 2 per thread).
  cumS[tid]       = cum[b * L_ + tid];
  cumS[tid + 256] = cum[b * L_ + tid + 256];
  __syncthreads();

  const int total = totals[b];
  const float* srcBase = hidden + (size_t)b * L_ * D_;
  float*       dstBase = out    + (size_t)b * max_T * D_;

  int kinds[FPW];   // 0 = out of grid, 1 = padding (zeros), 2 = gather
  const int t0 = tileBase + w;

  // ---- Phase 1: search + issue all async global->LDS loads --------------
  #pragma unroll
  for (int k = 0; k < FPW; ++k) {
    const int t = t0 + (k << 3);
    int kind = 0;
    if (t < max_T) kind = (t < total) ? 2 : 1;
    kinds[k] = kind;
    if (kind == 2) {
      // searchsorted(cum, t, side='right'): first i with cum[i] > t.
      int lo = 0, hi = L_;
      while (lo < hi) {
        const int mid = (lo + hi) >> 1;
        if (cumS[mid] <= t) lo = mid + 1; else hi = mid;
      }
      const int idx = (lo < L_) ? lo : (L_ - 1);
      const float4* src = (const float4*)(srcBase + (size_t)idx * D_);
      // 256 floats = 64 x float4; lane moves chunks [lane] and [lane+32].
      __builtin_amdgcn_global_load_async_to_lds_b128(
          (gv4i_p)(src + lane),      (lv4i_p)&stage[w][k][lane],      0, 0);
      __builtin_amdgcn_global_load_async_to_lds_b128(
          (gv4i_p)(src + lane + 32), (lv4i_p)&stage[w][k][lane + 32], 0, 0);
    }
  }

  // Drain all async loads before reading the staged rows back out.
  asm volatile("s_wait_asynccnt 0x0" ::: "memory");

  // ---- Phase 2: issue all async LDS->global stores (or zero padding) ----
  #pragma unroll
  for (int k = 0; k < FPW; ++k) {
    const int t = t0 + (k << 3);
    if (kinds[k] == 2) {
      float4* dst = (float4*)(dstBase + (size_t)t * D_);
      __builtin_amdgcn_global_store_async_from_lds_b128(
          (gv4i_p)(dst + lane),      (lv4i_p)&stage[w][k][lane],      0, 0);
      __builtin_amdgcn_global_store_async_from_lds_b128(
          (gv4i_p)(dst + lane + 32), (lv4i_p)&stage[w][k][lane + 32], 0, 0);
    } else if (kinds[k] == 1) {
      // Padding frames: write-once zeros, never re-read -> non-temporal b128
      // stores through a native vector type (NT builtin rejects HIP float4).
      v4i* dst = (v4i*)(dstBase + (size_t)t * D_);
      const v4i z = {0, 0, 0, 0};
      __builtin_nontemporal_store(z, dst + lane);
      __builtin_nontemporal_store(z, dst + lane + 32);
    }
  }
  // S_ENDPGM performs an implicit wait-idle, draining outstanding async stores.
}

// ---------------------------------------------------------------------------
// Scan kernel: per-batch inclusive prefix sum of durations (L=512) + totals.
// One block (256 threads) per batch; 2 elements per thread; LDS scan of the
// 256 per-thread partials. Negligible cost (32 tiny blocks).
// ---------------------------------------------------------------------------
__global__ __launch_bounds__(256) void lr_scan_kernel(
    const int* __restrict__ dur, int* __restrict__ cum, int* __restrict__ totals) {
  __shared__ int partial[256];
  const int b = blockIdx.x;
  const int tid = threadIdx.x;

  const int d0 = dur[b * L_ + 2 * tid];
  const int d1 = dur[b * L_ + 2 * tid + 1];
  const int sum2 = d0 + d1;
  partial[tid] = sum2;
  __syncthreads();

  // Hillis-Steele inclusive scan over 256 partials (8 steps).
  #pragma unroll
  for (int off = 1; off < 256; off <<= 1) {
    const int v = partial[tid];
    const int add = (tid >= off) ? partial[tid - off] : 0;
    __syncthreads();
    partial[tid] = v + add;
    __syncthreads();
  }

  const int incl = partial[tid];       // inclusive sum of this thread's pair
  const int excl = incl - sum2;        // exclusive prefix before this pair
  cum[b * L_ + 2 * tid]     = excl + d0;
  cum[b * L_ + 2 * tid + 1] = excl + sum2;
  if (tid == 255) totals[b] = incl;
}

// ---------------------------------------------------------------------------
extern "C" void kernel_launch(void* const* d_in, const int* in_sizes, int n_in,
                              void* d_out, int out_size, void* d_ws, size_t ws_size,
                              hipStream_t stream) {
  const float* hidden = (const float*)d_in[0];   // [B, L, D] float32
  const int*   dur    = (const int*)d_in[1];     // [B, L] int32
  float*       out    = (float*)d_out;           // [B, max_T, D] float32

  int* cum    = (int*)d_ws;                      // B*L ints = 64 KB
  int* totals = cum + B_ * L_;                   // B ints

  // max_T is data-dependent in the reference, but the harness sized d_out
  // with it: out_size = B * max_T * D.
  const int max_T = out_size / (B_ * D_);

  lr_scan_kernel<<<B_, 256, 0, stream>>>(dur, cum, totals);

  if (max_T > 0) {
    dim3 grid((max_T + T_TILE - 1) / T_TILE, B_);
    lr_gather_kernel<<<grid, 256, 0, stream>>>(hidden, cum, totals, out, max_T);
  }
}